// ChannelPatchEmbed_73512660238363
// MI455X (gfx1250) — compile-verified
//
#include <hip/hip_runtime.h>

// CDNA5 / gfx1250 wave32 WMMA patch-embed:
//   out[img][e][hp][wp] = sum_k W[e][k] * x[img][4hp+k/4][4wp+k%4] + bias[e]
// GEMM M=embeds(96) x N=patches x K=16 via V_WMMA_F32_16X16X4_F32.

typedef float v2f __attribute__((ext_vector_type(2)));
typedef float v8f __attribute__((ext_vector_type(8)));

#define IMG   512
#define HP    128      // patches per image edge
#define EMBED 96
#define NT    6        // 96 / 16 embed tiles
#define KS    4        // K=16 in steps of 4

__global__ __launch_bounds__(256)
void patch_embed_wmma_f32(const float* __restrict__ rgb,   // [8][3][512][512]
                          const float* __restrict__ hs,    // [8][4][512][512]
                          const float* __restrict__ dem,   // [8][1][512][512]
                          const float* __restrict__ W,     // [96][16]
                          const float* __restrict__ bias,  // [96]
                          float* __restrict__ out)         // [64][96][128][128]
{
    const int lane = threadIdx.x & 31;
    const int wave = threadIdx.x >> 5;
    const int tile = blockIdx.x * 8 + wave;      // 0 .. 65535 wave tiles

    const int wp0 = (tile & 7) << 4;             // patch-col base (16 patches/wave)
    const int hp  = (tile >> 3) & 127;           // patch row
    const int img = tile >> 10;                  // 0..63  (= b*8 + c)
    const int b   = img >> 3;
    const int c   = img & 7;

    // channel-concat [rgb(3) | hs(4) | dem(1)] folded into batch
    const float* src;
    if (c < 3)      src = rgb + (size_t)(b * 3 + c)       * (IMG * IMG);
    else if (c < 7) src = hs  + (size_t)(b * 4 + (c - 3)) * (IMG * IMG);
    else            src = dem + (size_t)b                 * (IMG * IMG);

    const int lhalf = lane >> 4;   // 0: lanes 0-15, 1: lanes 16-31
    const int llo   = lane & 15;

    // ---- A fragments: weight tiles, M=embed row, K=4 per step ----
    // f32 16x4 A layout: lanes 0-15 hold K={0,1}, lanes 16-31 hold K={2,3}.
    v2f afrag[NT][KS];
#pragma unroll
    for (int n = 0; n < NT; ++n) {
#pragma unroll
        for (int kk = 0; kk < KS; ++kk) {
            afrag[n][kk] = *(const v2f*)(W + (size_t)(16 * n + llo) * 16 + kk * 4 + 2 * lhalf);
        }
    }

    // ---- C init = bias (D layout: VGPR v, lane -> M = v + 8*lhalf) ----
    v8f acc[NT];
#pragma unroll
    for (int n = 0; n < NT; ++n) {
        acc[n] = *(const v8f*)(bias + 16 * n + 8 * lhalf);
    }

    // ---- K loop: B fragment = 16 patches x 4 k (pixel row kk of each patch) ----
    // B 4x16 layout mirrors A: lanes 0-15 K={0,1}, lanes 16-31 K={2,3}; N = lane%16.
    const int col = (wp0 + llo) * 4 + 2 * lhalf;
#pragma unroll
    for (int kk = 0; kk < KS; ++kk) {
        v2f bfrag = *(const v2f*)(src + (size_t)(hp * 4 + kk) * IMG + col);
#pragma unroll
        for (int n = 0; n < NT; ++n) {
            // 8 args: (neg_a, A, neg_b, B, c_mod, C, reuse_a, reuse_b)
            acc[n] = __builtin_amdgcn_wmma_f32_16x16x4_f32(
                false, afrag[n][kk], false, bfrag, (short)0, acc[n], false, false);
        }
    }

    // ---- store D: lane -> embed e = 16n + v + 8*lhalf, patch wp = wp0 + llo ----
    // lanes 0-15 and 16-31 each write a contiguous 64B run per VGPR.
    const size_t obase = (size_t)img * EMBED * (HP * HP) + (size_t)hp * HP + wp0 + llo;
#pragma unroll
    for (int n = 0; n < NT; ++n) {
#pragma unroll
        for (int v = 0; v < 8; ++v) {
            out[obase + (size_t)(16 * n + v + 8 * lhalf) * (HP * HP)] = acc[n][v];
        }
    }
}

extern "C" void kernel_launch(void* const* d_in, const int* in_sizes, int n_in,
                              void* d_out, int out_size, void* d_ws, size_t ws_size,
                              hipStream_t stream) {
    const float* rgb  = (const float*)d_in[0];
    const float* hs   = (const float*)d_in[1];
    const float* dem  = (const float*)d_in[2];
    const float* W    = (const float*)d_in[3];
    const float* bias = (const float*)d_in[4];
    float* out = (float*)d_out;

    // 64 imgs * 128 rows * 8 wp-tiles = 65536 wave tiles, 8 waves / 256-thread block
    dim3 grid(8192), block(256);
    patch_embed_wmma_f32<<<grid, block, 0, stream>>>(rgb, hs, dem, W, bias, out);
}